// MoPro_64536178589809
// MI455X (gfx1250) — compile-verified
//
#include <hip/hip_runtime.h>
#include <math.h>

typedef float v2f __attribute__((ext_vector_type(2)));
typedef float v8f __attribute__((ext_vector_type(8)));

#define NN 4096
#define DD 128
#define CC 100
#define KK 65536
#define INV_TEMP (1.0f/0.07f)
#define PROTO_M 0.999f
#define EPSN 1e-12f

// d_out flat layout (return order): pred, loss, logits, logits_proto, new_queue, new_protos
#define PRED_OFF   0LL
#define LOSS_OFF   409600LL
#define LOGITS_OFF 409601LL                       // 4096 x 65537
#define PROTOL_OFF 268849153LL                    // 4096 x 100
#define QUEUE_OFF  269258753LL                    // 128 x 65536
#define PROTOS_OFF 277647361LL                    // 100 x 128

// ---------------------------------------------------------------------------
// 1) l_neg = (q @ queue) / TEMP  via V_WMMA_F32_16X16X4_F32
//    grid: 16384 blocks x 256 thr (8 waves). wave -> one 16-row tile,
//    8 consecutive 16-col tiles. A (16x128 of q) resident in 64 VGPRs.
// ---------------------------------------------------------------------------
__global__ __launch_bounds__(256)
void gemm_lneg_kernel(const float* __restrict__ q,
                      const float* __restrict__ queue,
                      float* __restrict__ out) {
  const int lane = threadIdx.x & 31;
  const int wave = threadIdx.x >> 5;
  const int hi   = lane >> 4;          // K-half select (ISA f32 A/B layout)
  const int ln   = lane & 15;          // M (for A) / N (for B,C,D) across lanes
  const int CCHUNKS = (KK / 16) / 8;   // 512 column chunks
  const int mgroup  = blockIdx.x / CCHUNKS;   // 0..31
  const int cchunk  = blockIdx.x % CCHUNKS;
  const int mtile   = mgroup * 8 + wave;      // 0..255
  const int arow    = mtile * 16 + ln;

  // Preload A: lane holds q[arow][4i+2*hi .. +1] for i=0..31
  v2f areg[32];
  const float* qrow = q + (size_t)arow * DD;
  #pragma unroll
  for (int i = 0; i < 32; ++i) {
    const int d = 4 * i + 2 * hi;
    areg[i].x = qrow[d];
    areg[i].y = qrow[d + 1];
  }

  float* logits = out + LOGITS_OFF;

  for (int ct = 0; ct < 8; ++ct) {
    const int col0 = (cchunk * 8 + ct) * 16;
    const int col  = col0 + ln;
    v8f acc = {};
    #pragma unroll
    for (int i = 0; i < 32; ++i) {
      const int d = 4 * i + 2 * hi;
      v2f b;
      b.x = queue[(size_t)d       * KK + col];
      b.y = queue[(size_t)(d + 1) * KK + col];
      acc = __builtin_amdgcn_wmma_f32_16x16x4_f32(false, areg[i], false, b,
                                                  (short)0, acc, false, false);
    }
    // D layout: lane, vgpr v -> M = v + 8*hi, N = ln
    #pragma unroll
    for (int v = 0; v < 8; ++v) {
      const int rowM = mtile * 16 + v + 8 * hi;
      logits[(size_t)rowM * (KK + 1) + 1 + col] = acc[v] * INV_TEMP;
    }
  }
}

// ---------------------------------------------------------------------------
// 2) logits[:,0] = dot(q[n], k[n]) / TEMP
// ---------------------------------------------------------------------------
__global__ __launch_bounds__(128)
void lpos_kernel(const float* __restrict__ q, const float* __restrict__ k,
                 float* __restrict__ out) {
  __shared__ float s[128];
  const int n = blockIdx.x, t = threadIdx.x;
  s[t] = q[(size_t)n * DD + t] * k[(size_t)n * DD + t];
  __syncthreads();
  for (int off = 64; off > 0; off >>= 1) {
    if (t < off) s[t] += s[t + off];
    __syncthreads();
  }
  if (t == 0) out[LOGITS_OFF + (size_t)n * (KK + 1)] = s[0] * INV_TEMP;
}

// ---------------------------------------------------------------------------
// 3) logits_proto = (q @ prototypes^T) / TEMP   (4096 x 100)
// ---------------------------------------------------------------------------
__global__ __launch_bounds__(128)
void protologits_kernel(const float* __restrict__ q,
                        const float* __restrict__ protos,
                        float* __restrict__ out) {
  __shared__ float qs[128];
  const int n = blockIdx.x;
  qs[threadIdx.x] = q[(size_t)n * DD + threadIdx.x];
  __syncthreads();
  const int c = threadIdx.x;
  if (c < CC) {
    const float* pr = protos + (size_t)c * DD;
    float acc = 0.f;
    #pragma unroll 4
    for (int i = 0; i < DD; ++i) acc += qs[i] * pr[i];
    out[PROTOL_OFF + (size_t)n * CC + c] = acc * INV_TEMP;
  }
}

// ---------------------------------------------------------------------------
// 4) per-row instance loss: ws[n] = max + log(sum exp(x-max)) - x0
// ---------------------------------------------------------------------------
__global__ __launch_bounds__(256)
void inst_loss_kernel(const float* __restrict__ dout, float* __restrict__ ws_inst) {
  const float* row = dout + LOGITS_OFF + (size_t)blockIdx.x * (KK + 1);
  __shared__ float s[256];
  const int t = threadIdx.x;
  float m = -INFINITY;
  for (int i = t; i < KK + 1; i += 256) m = fmaxf(m, row[i]);
  s[t] = m; __syncthreads();
  for (int off = 128; off > 0; off >>= 1) {
    if (t < off) s[t] = fmaxf(s[t], s[t + off]);
    __syncthreads();
  }
  const float rmax = s[0]; __syncthreads();
  float sum = 0.f;
  for (int i = t; i < KK + 1; i += 256) sum += __expf(row[i] - rmax);
  s[t] = sum; __syncthreads();
  for (int off = 128; off > 0; off >>= 1) {
    if (t < off) s[t] += s[t + off];
    __syncthreads();
  }
  if (t == 0) ws_inst[blockIdx.x] = rmax + logf(s[0]) - row[0];
}

// ---------------------------------------------------------------------------
// 5) soft CE per row for (output,labels) and (logits_proto,labels)
// ---------------------------------------------------------------------------
__device__ float row_soft_ce(const float* __restrict__ xrow,
                             const float* __restrict__ lrow, float* s) {
  const int t = threadIdx.x;
  const float x = (t < CC) ? xrow[t] : -INFINITY;
  s[t] = x; __syncthreads();
  for (int off = 64; off > 0; off >>= 1) {
    if (t < off) s[t] = fmaxf(s[t], s[t + off]);
    __syncthreads();
  }
  const float m = s[0]; __syncthreads();
  s[t] = (t < CC) ? __expf(x - m) : 0.f; __syncthreads();
  for (int off = 64; off > 0; off >>= 1) {
    if (t < off) s[t] += s[t + off];
    __syncthreads();
  }
  const float lse = m + logf(s[0]); __syncthreads();
  s[t] = (t < CC) ? lrow[t] * (x - lse) : 0.f; __syncthreads();
  for (int off = 64; off > 0; off >>= 1) {
    if (t < off) s[t] += s[t + off];
    __syncthreads();
  }
  const float r = -s[0]; __syncthreads();
  return r;
}

__global__ __launch_bounds__(128)
void ce_rows_kernel(const float* __restrict__ output,
                    const float* __restrict__ labels,
                    const float* __restrict__ dout,
                    float* __restrict__ ws_cls, float* __restrict__ ws_proto) {
  __shared__ float s[128];
  const int n = blockIdx.x;
  const float* lrow = labels + (size_t)n * CC;
  float cls = row_soft_ce(output + (size_t)n * CC, lrow, s);
  float pro = row_soft_ce(dout + PROTOL_OFF + (size_t)n * CC, lrow, s);
  if (threadIdx.x == 0) { ws_cls[n] = cls; ws_proto[n] = pro; }
}

// ---------------------------------------------------------------------------
// 6) pred = get_label(output, 1); col 0 forced to (rowsum(mask)==0)
// ---------------------------------------------------------------------------
__global__ __launch_bounds__(128)
void pred_kernel(const float* __restrict__ output, float* __restrict__ out) {
  __shared__ float s[128];
  __shared__ int cnt[128];
  const int n = blockIdx.x, t = threadIdx.x;
  const float x = (t < CC) ? output[(size_t)n * CC + t] : -INFINITY;
  s[t] = x; __syncthreads();
  for (int off = 64; off > 0; off >>= 1) {
    if (t < off) s[t] = fmaxf(s[t], s[t + off]);
    __syncthreads();
  }
  const float m = s[0]; __syncthreads();
  const int mask = (t < CC && x >= m) ? 1 : 0;
  cnt[t] = mask; __syncthreads();
  for (int off = 64; off > 0; off >>= 1) {
    if (t < off) cnt[t] += cnt[t + off];
    __syncthreads();
  }
  const int total = cnt[0];
  if (t < CC)
    out[PRED_OFF + (size_t)n * CC + t] =
        (t == 0) ? ((total == 0) ? 1.f : 0.f) : (float)mask;
}

// ---------------------------------------------------------------------------
// 7) new_queue = queue with k^T spliced at columns [ptr, ptr+N)
// ---------------------------------------------------------------------------
__global__ __launch_bounds__(256)
void queue_kernel(const float* __restrict__ queue, const float* __restrict__ k,
                  const int* __restrict__ ptr, float* __restrict__ out) {
  const size_t idx = (size_t)blockIdx.x * 256 + threadIdx.x;
  const int col = (int)(idx & (KK - 1));
  const int d   = (int)(idx >> 16);
  const int p   = ptr[0];
  float v;
  if (col >= p && col < p + NN) v = k[(size_t)(col - p) * DD + d];
  else                          v = queue[idx];
  out[QUEUE_OFF + idx] = v;
}

// ---------------------------------------------------------------------------
// 8) order-faithful per-class prototype EMA + row L2 norm
// ---------------------------------------------------------------------------
__global__ __launch_bounds__(128)
void proto_ema_kernel(const float* __restrict__ protos, const float* __restrict__ q,
                      const int* __restrict__ targets, float* __restrict__ out) {
  __shared__ float s[128];
  const int c = blockIdx.x, d = threadIdx.x;
  float p = protos[(size_t)c * DD + d];
  if (c != 0) {
    for (int n = 0; n < NN; ++n) {
      if (targets[n] == c)
        p = p * PROTO_M + (1.f - PROTO_M) * q[(size_t)n * DD + d];
    }
  }
  s[d] = p * p; __syncthreads();
  for (int off = 64; off > 0; off >>= 1) {
    if (d < off) s[d] += s[d + off];
    __syncthreads();
  }
  const float nrm = fmaxf(sqrtf(s[0]), EPSN);
  out[PROTOS_OFF + (size_t)c * DD + d] = p / nrm;
}

// ---------------------------------------------------------------------------
// 9) loss = (sum ws_inst + sum ws_cls + sum ws_proto) / N   (weights = 1)
// ---------------------------------------------------------------------------
__global__ __launch_bounds__(256)
void loss_kernel(const float* __restrict__ ws, float* __restrict__ out) {
  __shared__ float s[256];
  const int t = threadIdx.x;
  float a = 0.f;
  for (int i = t; i < 3 * NN; i += 256) a += ws[i];
  s[t] = a; __syncthreads();
  for (int off = 128; off > 0; off >>= 1) {
    if (t < off) s[t] += s[t + off];
    __syncthreads();
  }
  if (t == 0) out[LOSS_OFF] = s[0] / (float)NN;
}

extern "C" void kernel_launch(void* const* d_in, const int* in_sizes, int n_in,
                              void* d_out, int out_size, void* d_ws, size_t ws_size,
                              hipStream_t stream) {
  const float* q       = (const float*)d_in[0];
  const float* k       = (const float*)d_in[1];
  const float* output  = (const float*)d_in[2];
  const float* queue   = (const float*)d_in[3];
  const float* protos  = (const float*)d_in[4];
  const float* labels  = (const float*)d_in[5];
  const int*   targets = (const int*)d_in[6];
  const int*   ptr     = (const int*)d_in[7];
  float* out = (float*)d_out;
  float* ws  = (float*)d_ws;            // [0,N): inst  [N,2N): cls  [2N,3N): proto

  gemm_lneg_kernel<<<16384, 256, 0, stream>>>(q, queue, out);
  lpos_kernel<<<NN, 128, 0, stream>>>(q, k, out);
  protologits_kernel<<<NN, 128, 0, stream>>>(q, protos, out);
  inst_loss_kernel<<<NN, 256, 0, stream>>>(out, ws);
  ce_rows_kernel<<<NN, 128, 0, stream>>>(output, labels, out, ws + NN, ws + 2 * NN);
  pred_kernel<<<NN, 128, 0, stream>>>(output, out);
  queue_kernel<<<(DD * KK) / 256, 256, 0, stream>>>(queue, k, ptr, out);
  proto_ema_kernel<<<CC, 128, 0, stream>>>(protos, q, targets, out);
  loss_kernel<<<1, 256, 0, stream>>>(ws, out);
}